// TreeGATLayer_77970836291712
// MI455X (gfx1250) — compile-verified
//
#include <hip/hip_runtime.h>
#include <hip/hip_bf16.h>

#define BB  4
#define LL  2048
#define DD  256
#define HH  4
#define DHH 64
#define NROWS (BB * LL)   // 8192
#define EPSV 1e-5f

#define BLK_M 128         // 8 waves x 16 rows
#define BLK_N 64
#define KPAD  260         // 256 + 4 dword pad: distinct LDS banks for B-frag reads

typedef float v2f __attribute__((ext_vector_type(2)));
typedef float v8f __attribute__((ext_vector_type(8)));
typedef int   v4i __attribute__((vector_size(4 * sizeof(int))));
typedef __attribute__((address_space(1))) v4i* global_v4i_ptr;
typedef __attribute__((address_space(3))) v4i* lds_v4i_ptr;

#if defined(__has_builtin)
#if __has_builtin(__builtin_amdgcn_global_load_async_to_lds_b128) && \
    __has_builtin(__builtin_amdgcn_s_wait_asynccnt)
#define USE_ASYNC_LDS 1
#endif
#endif

__device__ __forceinline__ float wave_sum32(float v) {
#pragma unroll
    for (int off = 16; off > 0; off >>= 1) v += __shfl_xor(v, off, 32);
    return v;
}

// -------------------- LayerNorm: one wave per row of 256 --------------------
__global__ void ln_kernel(const float* __restrict__ x,
                          const float* __restrict__ gamma,
                          const float* __restrict__ beta,
                          float* __restrict__ h) {
    const int wave = (int)((blockIdx.x * blockDim.x + threadIdx.x) >> 5);
    const int lane = (int)(threadIdx.x & 31);
    if (wave >= NROWS) return;
    const float* row = x + (size_t)wave * DD;
    float vals[8];
    float s = 0.f;
#pragma unroll
    for (int t = 0; t < 8; ++t) { vals[t] = row[lane + 32 * t]; s += vals[t]; }
    s = wave_sum32(s);
    const float mu = s * (1.0f / DD);
    float sq = 0.f;
#pragma unroll
    for (int t = 0; t < 8; ++t) { float d = vals[t] - mu; sq += d * d; }
    sq = wave_sum32(sq);
    const float rstd = rsqrtf(sq * (1.0f / DD) + EPSV);
    float* orow = h + (size_t)wave * DD;
#pragma unroll
    for (int t = 0; t < 8; ++t) {
        const int col = lane + 32 * t;
        orow[col] = (vals[t] - mu) * rstd * gamma[col] + beta[col];
    }
}

// ---- Blocked NT-GEMM, f32 WMMA: out[M,N] = A[M,K] @ W[N,K]^T + bias (+resid)
// Block: 256 threads (8 waves) -> 128x64 tile; W slab staged in LDS (async),
// K chunked by 16 (4 a-frags + 16 b-frags batched, then 16 WMMAs).
__global__ __launch_bounds__(256) void wmma_gemm_nt(
        const float* __restrict__ A,
        const float* __restrict__ W,
        const float* __restrict__ bias,
        const float* __restrict__ resid,
        float* __restrict__ out,
        int M, int N, int K) {
    __shared__ float Ws[BLK_N * KPAD];   // ~65 KB
    const int tid  = (int)threadIdx.x;
    const int lane = tid & 31;
    const int wid  = tid >> 5;           // 0..7
    const int tilesN = N / BLK_N;
    const int bm = (int)blockIdx.x / tilesN;
    const int bn = (int)blockIdx.x - bm * tilesN;
    const int row0 = bm * BLK_M + wid * 16;
    const int col0 = bn * BLK_N;

    // ---- stage W[col0 .. col0+63][0..K) into LDS (16B per lane per iter) ----
#pragma unroll
    for (int it = 0; it < (BLK_N * 64) / 256; ++it) {   // 4096 float4s / 256 thr
        const int f  = it * 256 + tid;                  // float4 index
        const int n  = f >> 6;                          // 0..63
        const int k4 = (f & 63) << 2;                   // 0..252 step 4
        const float* src = W + (size_t)(col0 + n) * K + k4;
        float* dst = &Ws[n * KPAD + k4];
#if defined(USE_ASYNC_LDS)
        __builtin_amdgcn_global_load_async_to_lds_b128(
            (global_v4i_ptr)src, (lds_v4i_ptr)dst, 0, 0);
#else
        *(float4*)dst = *(const float4*)src;
#endif
    }
#if defined(USE_ASYNC_LDS)
    __builtin_amdgcn_s_wait_asynccnt(0);
#endif
    __syncthreads();

    // ---- fragment bases ----
    const int mrow = lane & 15;
    const int koff = (lane >> 4) << 1;   // 0 or 2
    const float* a_base = A + (size_t)(row0 + mrow) * K + koff;
    const float* b_base = &Ws[mrow * KPAD + koff];

    v8f acc0 = {}, acc1 = {}, acc2 = {}, acc3 = {};
    for (int k = 0; k < K; k += 16) {
        v2f a[4], bf[4][4];
#pragma unroll
        for (int s = 0; s < 4; ++s)
            a[s] = *(const v2f*)(a_base + k + 4 * s);
#pragma unroll
        for (int c = 0; c < 4; ++c)
#pragma unroll
            for (int s = 0; s < 4; ++s)
                bf[c][s] = *(const v2f*)(b_base + c * 16 * KPAD + k + 4 * s);
#pragma unroll
        for (int s = 0; s < 4; ++s) {
            acc0 = __builtin_amdgcn_wmma_f32_16x16x4_f32(false, a[s], false, bf[0][s], (short)0, acc0, false, false);
            acc1 = __builtin_amdgcn_wmma_f32_16x16x4_f32(false, a[s], false, bf[1][s], (short)0, acc1, false, false);
            acc2 = __builtin_amdgcn_wmma_f32_16x16x4_f32(false, a[s], false, bf[2][s], (short)0, acc2, false, false);
            acc3 = __builtin_amdgcn_wmma_f32_16x16x4_f32(false, a[s], false, bf[3][s], (short)0, acc3, false, false);
        }
    }

    // D layout: VGPR e -> row = row0 + e + (lane>=16 ? 8 : 0), col = col0 + c*16 + (lane&15)
    const int ocol = lane & 15;
    const int orow = row0 + ((lane >> 4) << 3);
    const float bb0 = bias[col0 + ocol];
    const float bb1 = bias[col0 + 16 + ocol];
    const float bb2 = bias[col0 + 32 + ocol];
    const float bb3 = bias[col0 + 48 + ocol];
#pragma unroll
    for (int e = 0; e < 8; ++e) {
        const size_t base = (size_t)(orow + e) * N + col0 + ocol;
        float r0 = acc0[e] + bb0;
        float r1 = acc1[e] + bb1;
        float r2 = acc2[e] + bb2;
        float r3 = acc3[e] + bb3;
        if (resid) {
            r0 += resid[base];
            r1 += resid[base + 16];
            r2 += resid[base + 32];
            r3 += resid[base + 48];
        }
        out[base]      = r0;
        out[base + 16] = r1;
        out[base + 32] = r2;
        out[base + 48] = r3;
    }
}

// -------- Sparse tree attention: one wave per (b,i), all 4 heads --------
__global__ void tree_attn_kernel(const float* __restrict__ q,
                                 const float* __restrict__ k,
                                 const float* __restrict__ v,
                                 const int* __restrict__ parents,
                                 const unsigned char* __restrict__ pad,
                                 float* __restrict__ y) {
    const int wave = (int)((blockIdx.x * blockDim.x + threadIdx.x) >> 5);
    const int lane = (int)(threadIdx.x & 31);
    if (wave >= NROWS) return;
    const int b = wave / LL;
    const int i = wave - b * LL;

    const size_t basei = (size_t)wave * DD;
    float qv[HH][2];
#pragma unroll
    for (int hh = 0; hh < HH; ++hh) {
        qv[hh][0] = q[basei + hh * DHH + lane];
        qv[hh][1] = q[basei + hh * DHH + lane + 32];
    }
    float m[HH], s[HH], acc[HH][2];
#pragma unroll
    for (int hh = 0; hh < HH; ++hh) {
        m[hh] = -3.4e38f; s[hh] = 0.f; acc[hh][0] = 0.f; acc[hh][1] = 0.f;
    }

    const int pi = parents[wave];
    const bool vpi = (pi >= 0) && (pi < LL);
    const bool padi = pad[wave] != 0;

    for (int j0 = 0; j0 < LL; j0 += 32) {
        const int j = j0 + lane;
        const int pj = parents[(size_t)b * LL + j];
        const bool padj = pad[(size_t)b * LL + j] != 0;
        const bool adj = padi && padj &&
                         ((j == i) || (vpi && pi == j) ||
                          ((pj >= 0) && (pj < LL) && (pj == i)));
        unsigned mask = __builtin_amdgcn_ballot_w32(adj);
        while (mask) {
            const int l2 = __ffs(mask) - 1;
            mask &= mask - 1;
            const int jj = j0 + l2;
            const size_t basej = ((size_t)b * LL + jj) * DD;
#pragma unroll
            for (int hh = 0; hh < HH; ++hh) {
                const float k0 = k[basej + hh * DHH + lane];
                const float k1 = k[basej + hh * DHH + lane + 32];
                float prod = qv[hh][0] * k0 + qv[hh][1] * k1;
                prod = wave_sum32(prod);
                const float logit = prod * 0.125f;  // 1/sqrt(64)
                const float nm = fmaxf(m[hh], logit);
                const float sc = __expf(m[hh] - nm);
                const float w = __expf(logit - nm);
                s[hh] = s[hh] * sc + w;
                acc[hh][0] = acc[hh][0] * sc + w * v[basej + hh * DHH + lane];
                acc[hh][1] = acc[hh][1] * sc + w * v[basej + hh * DHH + lane + 32];
                m[hh] = nm;
            }
        }
    }
#pragma unroll
    for (int hh = 0; hh < HH; ++hh) {
        const float inv = (s[hh] > 0.f) ? (1.0f / s[hh]) : 0.f;
        y[basei + hh * DHH + lane]      = acc[hh][0] * inv;
        y[basei + hh * DHH + lane + 32] = acc[hh][1] * inv;
    }
}

extern "C" void kernel_launch(void* const* d_in, const int* in_sizes, int n_in,
                              void* d_out, int out_size, void* d_ws, size_t ws_size,
                              hipStream_t stream) {
    const float*         x       = (const float*)d_in[0];
    const int*           parents = (const int*)d_in[1];
    const unsigned char* pad     = (const unsigned char*)d_in[2];
    const float* Wq = (const float*)d_in[3];
    const float* bq = (const float*)d_in[4];
    const float* Wk = (const float*)d_in[5];
    const float* bk = (const float*)d_in[6];
    const float* Wv = (const float*)d_in[7];
    const float* bv = (const float*)d_in[8];
    const float* Wo = (const float*)d_in[9];
    const float* bo = (const float*)d_in[10];
    const float* gamma = (const float*)d_in[11];
    const float* beta  = (const float*)d_in[12];
    float* out = (float*)d_out;

    const size_t mat = (size_t)NROWS * DD;  // 2M floats = 8MB
    float* h  = (float*)d_ws;               // layernorm out; reused as attn out y
    float* qb = h + mat;
    float* kb = qb + mat;
    float* vb = kb + mat;

    const int ln_blocks   = NROWS / 4;                          // 128 thr, 4 waves
    const int gemm_blocks = (NROWS / BLK_M) * (DD / BLK_N);     // 64*4 = 256
    const int attn_blocks = NROWS / 4;

    ln_kernel<<<ln_blocks, 128, 0, stream>>>(x, gamma, beta, h);

    wmma_gemm_nt<<<gemm_blocks, 256, 0, stream>>>(h, Wq, bq, nullptr, qb, NROWS, DD, DD);
    wmma_gemm_nt<<<gemm_blocks, 256, 0, stream>>>(h, Wk, bk, nullptr, kb, NROWS, DD, DD);
    wmma_gemm_nt<<<gemm_blocks, 256, 0, stream>>>(h, Wv, bv, nullptr, vb, NROWS, DD, DD);

    // y reuses h (h is dead after the three projections)
    tree_attn_kernel<<<attn_blocks, 128, 0, stream>>>(qb, kb, vb, parents, pad, h);

    wmma_gemm_nt<<<gemm_blocks, 256, 0, stream>>>(h, Wo, bo, x, out, NROWS, DD, DD);
}